// Caption_83236466196715
// MI455X (gfx1250) — compile-verified
//
#include <hip/hip_runtime.h>
#include <math.h>

// ---------------- problem constants ----------------
#define BN   64     // batch
#define TT   24     // timesteps
#define MM   512    // conv out channels / context dim
#define VV   12000  // vocab
#define CH   2048   // image feature channels
#define EE   196    // embedding / hidden size
#define SS   196    // spatial (14*14)
#define G4   784    // 4*EE gates
#define KX0  708    // EE + MM   (layer0 input)
#define KX1  392    // 2*EE      (layer1 input)
#define KLIN 904    // MM + 2*EE (lin input)

typedef float v2f __attribute__((ext_vector_type(2)));
typedef float v8f __attribute__((ext_vector_type(8)));

__device__ __forceinline__ v8f wmma_k4(v2f a, v2f b, v8f c) {
  // D(16x16) = A(16x4) * B(4x16) + C, full fp32
  return __builtin_amdgcn_wmma_f32_16x16x4_f32(false, a, false, b, (short)0, c,
                                               false, false);
}

__device__ __forceinline__ v2f lrelu2(v2f a) {
  a.x = fmaxf(a.x, 0.f) + 0.01f * fminf(a.x, 0.f);
  a.y = fmaxf(a.y, 0.f) + 0.01f * fminf(a.y, 0.f);
  return a;
}

// ------------------------------------------------------------------
// 4 row-tiles x 16 cols per wave, software-pipelined: fragments for
// step k+4 are loaded before the 4 WMMAs of step k are issued, so the
// load waits overlap matrix-pipe execution.
// ------------------------------------------------------------------
__device__ __forceinline__ void mm4(const float* __restrict__ abase,
                                    size_t rstride,   // 16*lda
                                    const float* __restrict__ brow, int K,
                                    v8f& d0, v8f& d1, v8f& d2, v8f& d3)
{
  v2f b  = *(const v2f*)(brow);
  v2f x0 = *(const v2f*)(abase);
  v2f x1 = *(const v2f*)(abase + rstride);
  v2f x2 = *(const v2f*)(abase + 2 * rstride);
  v2f x3 = *(const v2f*)(abase + 3 * rstride);
#pragma unroll 4
  for (int k = 4; k < K; k += 4) {
    v2f bn = *(const v2f*)(brow + k);
    v2f y0 = *(const v2f*)(abase + k);
    v2f y1 = *(const v2f*)(abase + rstride + k);
    v2f y2 = *(const v2f*)(abase + 2 * rstride + k);
    v2f y3 = *(const v2f*)(abase + 3 * rstride + k);
    d0 = wmma_k4(x0, b, d0);
    d1 = wmma_k4(x1, b, d1);
    d2 = wmma_k4(x2, b, d2);
    d3 = wmma_k4(x3, b, d3);
    b = bn; x0 = y0; x1 = y1; x2 = y2; x3 = y3;
  }
  d0 = wmma_k4(x0, b, d0);
  d1 = wmma_k4(x1, b, d1);
  d2 = wmma_k4(x2, b, d2);
  d3 = wmma_k4(x3, b, d3);
}

// ------------------------------------------------------------------
// Tensor Data Mover: 2-D tile (w elems x h rows, 4-byte data,
// row stride `strd` elems) global -> LDS (compacted).
// ------------------------------------------------------------------
#if __has_builtin(__builtin_amdgcn_tensor_load_to_lds)
#define HAVE_TDM 1
typedef unsigned int tdm_v4u __attribute__((ext_vector_type(4)));
typedef int tdm_v4i __attribute__((ext_vector_type(4)));
typedef int tdm_v8i __attribute__((ext_vector_type(8)));

__device__ __forceinline__ void tdm_load_2d(void* lds, const void* g,
                                            unsigned w, unsigned h, unsigned strd)
{
  unsigned long long ga = (unsigned long long)g;
  tdm_v4u g0;
  g0[0] = 1u;                                   // count=1 (valid user D#)
  g0[1] = (unsigned)(size_t)lds;                // lds_addr (bytes)
  g0[2] = (unsigned)ga;                         // global_addr[31:0]
  g0[3] = (unsigned)((ga >> 32) & 0x1FFFFFFu) | 0x80000000u;  // [56:32] | type=2
  tdm_v8i g1;
  g1[0] = (int)(2u << 16);                                      // data_size=4B
  g1[1] = (int)((w & 0xFFFFu) << 16);                           // tensor_dim0 lo
  g1[2] = (int)(((w >> 16) & 0xFFFFu) | ((h & 0xFFFFu) << 16)); // dim0 hi|dim1 lo
  g1[3] = (int)(((h >> 16) & 0xFFFFu) | ((w & 0xFFFFu) << 16)); // dim1 hi|tile0
  g1[4] = (int)(h & 0xFFFFu);                                   // tile1 (tile2=0)
  g1[5] = (int)strd;                                            // dim0 stride lo
  g1[6] = 0;
  g1[7] = 0;
  tdm_v4i gz = {0, 0, 0, 0};
#if __clang_major__ >= 23
  tdm_v8i gz8 = {0, 0, 0, 0, 0, 0, 0, 0};
  __builtin_amdgcn_tensor_load_to_lds(g0, g1, gz, gz, gz8, 0);
#else
  __builtin_amdgcn_tensor_load_to_lds(g0, g1, gz, gz, 0);
#endif
}
#else
#define HAVE_TDM 0
#endif

// ------------------------------------------------------------------
// Generic OUT[r][c] = X[r][:] . W[c][:] + bias[c], rows fixed at 64.
// one wave per 16-col tile, 4 row tiles per wave. cols guarded.
// ------------------------------------------------------------------
__global__ __launch_bounds__(32) void gemm_xwT(
    const float* __restrict__ X, int ldx,
    const float* __restrict__ W, int ldw,
    const float* __restrict__ bias,
    float* __restrict__ OUT, int ldo,
    int K, int ncols)
{
  const int lane = threadIdx.x;
  const int hl = lane >> 4, ln = lane & 15;
  const int c  = blockIdx.x * 16 + ln;
  const int cc = c < ncols ? c : (ncols - 1);

  v8f a0 = {}, a1 = {}, a2 = {}, a3 = {};
  mm4(X + (size_t)ln * ldx + 2 * hl, (size_t)16 * ldx,
      W + (size_t)cc * ldw + 2 * hl, K, a0, a1, a2, a3);

  if (c < ncols) {
    const float bv = bias[c];
    const v8f* acc[4] = {&a0, &a1, &a2, &a3};
#pragma unroll
    for (int rt = 0; rt < 4; ++rt)
#pragma unroll
      for (int r = 0; r < 8; ++r)
        OUT[(size_t)(rt * 16 + r + 8 * hl) * ldo + c] = (*acc[rt])[r] + bv;
  }
}

// ------------------------------------------------------------------
// mapped[n][m][s] = img[n][s][:] . conv_w[m][:] + conv_b[m]
// wave: 64 m-rows x 16 s-cols (img B fragment reused 4x)
// grid: (13 s-tiles, 8 m-groups, 64 n)
// ------------------------------------------------------------------
__global__ __launch_bounds__(32) void conv1x1_gemm(
    const float* __restrict__ img,     // (BN, SS, CH)
    const float* __restrict__ conv_w,  // (MM, CH)
    const float* __restrict__ conv_b,  // (MM)
    float* __restrict__ mapped)        // (BN, MM, SS)
{
  const int lane = threadIdx.x;
  const int hl = lane >> 4, ln = lane & 15;
  const int s0 = blockIdx.x * 16;
  const int mg = blockIdx.y * 64;
  const int n  = blockIdx.z;

  const int s  = s0 + ln;
  const int sc = s < SS ? s : (SS - 1);

  v8f a0 = {}, a1 = {}, a2 = {}, a3 = {};
  mm4(conv_w + (size_t)(mg + ln) * CH + 2 * hl, (size_t)16 * CH,
      img + ((size_t)n * SS + sc) * CH + 2 * hl, CH, a0, a1, a2, a3);

  if (s < SS) {
    const v8f* acc[4] = {&a0, &a1, &a2, &a3};
#pragma unroll
    for (int rt = 0; rt < 4; ++rt)
#pragma unroll
      for (int r = 0; r < 8; ++r) {
        const int m = mg + rt * 16 + r + 8 * hl;
        mapped[((size_t)n * MM + m) * SS + s] = (*acc[rt])[r] + conv_b[m];
      }
  }
}

// ------------------------------------------------------------------
// gates[d][n][g] = x[n][:].w_ih[d][g][:] + h[d][n][:].w_hh[d][g][:] + b[d][g]
// grid: (49 col tiles, 2 dirs); wave covers all 64 batch rows.
// ------------------------------------------------------------------
__global__ __launch_bounds__(32) void lstm_gates(
    const float* __restrict__ x,  int Kx,     // (BN, Kx)
    const float* __restrict__ hs,             // (2, BN, EE)
    const float* __restrict__ w_ih,           // (2, G4, Kx)
    const float* __restrict__ w_hh,           // (2, G4, EE)
    const float* __restrict__ bias,           // (2, G4)
    float* __restrict__ gates)                // (2, BN, G4)
{
  const int lane = threadIdx.x;
  const int hl = lane >> 4, ln = lane & 15;
  const int c = blockIdx.x * 16 + ln;
  const int d = blockIdx.y;

  const float* h  = hs + (size_t)d * BN * EE;
  const float* wi = w_ih + (size_t)d * G4 * Kx;
  const float* wh = w_hh + (size_t)d * G4 * EE;

  v8f a0 = {}, a1 = {}, a2 = {}, a3 = {};
  mm4(x + (size_t)ln * Kx + 2 * hl, (size_t)16 * Kx,
      wi + (size_t)c * Kx + 2 * hl, Kx, a0, a1, a2, a3);
  mm4(h + (size_t)ln * EE + 2 * hl, (size_t)16 * EE,
      wh + (size_t)c * EE + 2 * hl, EE, a0, a1, a2, a3);

  const float bv = bias[(size_t)d * G4 + c];
  const v8f* acc[4] = {&a0, &a1, &a2, &a3};
#pragma unroll
  for (int rt = 0; rt < 4; ++rt)
#pragma unroll
    for (int r = 0; r < 8; ++r)
      gates[((size_t)d * BN + rt * 16 + r + 8 * hl) * G4 + c] = (*acc[rt])[r] + bv;
}

// ------------------------------------------------------------------
// Word projection: OUT = leaky_relu(A) @ W.T + bias, A = (64,512).
// Block = 192 threads (6 waves, 1 col-tile each, all 64 rows).
// A staged into LDS in two 64KB K-chunks by the Tensor Data Mover;
// waves read A fragments from LDS (ds_load_b64), W streams from L2.
// grid: 125 blocks (125*6 = 750 col tiles).
// ------------------------------------------------------------------
__global__ __launch_bounds__(192) void wp_gemm(
    const float* __restrict__ A,     // (64, 512)
    const float* __restrict__ W,     // (VV, 512)
    const float* __restrict__ bias,  // (VV)
    float* __restrict__ OUT)         // (64, VV)
{
  __shared__ float As[64 * 256];     // 64 KB chunk of A
  const int tid  = threadIdx.x;
  const int wave = tid >> 5;
  const int lane = tid & 31;
  const int hl = lane >> 4, ln = lane & 15;
  const int c = (blockIdx.x * 6 + wave) * 16 + ln;
  const float* wrow = W + (size_t)c * 512 + 2 * hl;

  v8f a0 = {}, a1 = {}, a2 = {}, a3 = {};

  for (int kb = 0; kb < 512; kb += 256) {
#if HAVE_TDM
    if (tid < 32) {                       // wave 0 drives the TDM
      tdm_load_2d(As, A + kb, 256u, 64u, 512u);
#if __has_builtin(__builtin_amdgcn_s_wait_tensorcnt)
      __builtin_amdgcn_s_wait_tensorcnt(0);
#else
      asm volatile("s_wait_tensorcnt 0x0" ::: "memory");
#endif
    }
#else
    for (int i = tid; i < 64 * 256; i += 192)
      As[i] = A[(size_t)(i >> 8) * 512 + kb + (i & 255)];
#endif
    __syncthreads();

    // software-pipelined inner loop over this 256-wide K chunk
    const float* arow = As + ln * 256 + 2 * hl;
    v2f b  = *(const v2f*)(wrow + kb);
    v2f x0 = lrelu2(*(const v2f*)(arow));
    v2f x1 = lrelu2(*(const v2f*)(arow + 16 * 256));
    v2f x2 = lrelu2(*(const v2f*)(arow + 32 * 256));
    v2f x3 = lrelu2(*(const v2f*)(arow + 48 * 256));
#pragma unroll 4
    for (int k = 4; k < 256; k += 4) {
      v2f bn = *(const v2f*)(wrow + kb + k);
      v2f y0 = lrelu2(*(const v2f*)(arow + k));
      v2f y1 = lrelu2(*(const v2f*)(arow + 16 * 256 + k));
      v2f y2 = lrelu2(*(const v2f*)(arow + 32 * 256 + k));
      v2f y3 = lrelu2(*(const v2f*)(arow + 48 * 256 + k));
      a0 = wmma_k4(x0, b, a0);
      a1 = wmma_k4(x1, b, a1);
      a2 = wmma_k4(x2, b, a2);
      a3 = wmma_k4(x3, b, a3);
      b = bn; x0 = y0; x1 = y1; x2 = y2; x3 = y3;
    }
    a0 = wmma_k4(x0, b, a0);
    a1 = wmma_k4(x1, b, a1);
    a2 = wmma_k4(x2, b, a2);
    a3 = wmma_k4(x3, b, a3);
    __syncthreads();   // before next chunk overwrites LDS
  }

  const float bv = bias[c];
  const v8f* acc[4] = {&a0, &a1, &a2, &a3};
#pragma unroll
  for (int rt = 0; rt < 4; ++rt)
#pragma unroll
    for (int r = 0; r < 8; ++r)
      OUT[(size_t)(rt * 16 + r + 8 * hl) * VV + c] = (*acc[rt])[r] + bv;
}

// ------------------------------------------------------------------
// LSTM elementwise cell; also writes h into a concat buffer
// ------------------------------------------------------------------
__global__ void lstm_cell(const float* __restrict__ gates,
                          float* __restrict__ h, float* __restrict__ c,
                          float* __restrict__ concat_out, int cstride)
{
  int idx = blockIdx.x * blockDim.x + threadIdx.x;
  if (idx >= 2 * BN * EE) return;
  const int j = idx % EE;
  const int n = (idx / EE) % BN;
  const int d = idx / (EE * BN);

  const float* gr = gates + ((size_t)d * BN + n) * G4;
  const float gi = gr[j];
  const float gf = gr[EE + j];
  const float gg = gr[2 * EE + j];
  const float go = gr[3 * EE + j];

  const float si = 1.f / (1.f + __expf(-gi));
  const float sf = 1.f / (1.f + __expf(-gf));
  const float so = 1.f / (1.f + __expf(-go));

  const size_t hidx = ((size_t)d * BN + n) * EE + j;
  const float cv = sf * c[hidx] + si * tanhf(gg);
  c[hidx] = cv;
  const float hv = so * tanhf(cv);
  h[hidx] = hv;
  concat_out[(size_t)n * cstride + d * EE + j] = hv;
}

// pooled[n][ch] = mean_s img[n][s][ch]
__global__ void pool_mean(const float* __restrict__ img,
                          float* __restrict__ pooled)
{
  int idx = blockIdx.x * blockDim.x + threadIdx.x;
  if (idx >= BN * CH) return;
  const int n = idx >> 11, ch = idx & (CH - 1);
  const float* p = img + (size_t)n * SS * CH + ch;
  float s = 0.f;
  for (int si = 0; si < SS; ++si) s += p[(size_t)si * CH];
  pooled[idx] = s * (1.f / (float)SS);
}

// tmp[n][m] = dot(mapped[n][m][:], v[n][:])   one wave per (n,m)
__global__ __launch_bounds__(256) void attn_dot(
    const float* __restrict__ mapped,
    const float* __restrict__ v,       // (BN, EE)
    float* __restrict__ outdot)        // (BN, MM)
{
  const int wid  = (blockIdx.x * blockDim.x + threadIdx.x) >> 5;
  const int lane = threadIdx.x & 31;
  if (wid >= BN * MM) return;
  const int n = wid >> 9, m = wid & (MM - 1);
  const float* row = mapped + ((size_t)n * MM + m) * SS;
  const float* vv  = v + (size_t)n * EE;
  float s = 0.f;
  for (int j = lane; j < SS; j += 32) s += row[j] * vv[j];
#pragma unroll
  for (int off = 16; off; off >>= 1) s += __shfl_down(s, off, 32);
  if (lane == 0) outdot[wid] = s;
}

// ctx[n][:] = tmp[n][:] / max(||tmp[n]||, 1e-12)   one block (256) per n
__global__ __launch_bounds__(256) void l2norm_row(
    const float* __restrict__ in, float* __restrict__ out)
{
  __shared__ float red[8];
  const int n = blockIdx.x, t = threadIdx.x;
  const float* r = in + (size_t)n * MM;
  const float a = r[t], b = r[t + 256];
  float ss = a * a + b * b;
#pragma unroll
  for (int off = 16; off; off >>= 1) ss += __shfl_down(ss, off, 32);
  if ((t & 31) == 0) red[t >> 5] = ss;
  __syncthreads();
  if (t < 8) {
    float x = red[t];
#pragma unroll
    for (int off = 4; off; off >>= 1) x += __shfl_down(x, off, 8);
    if (t == 0) red[0] = x;
  }
  __syncthreads();
  const float scale = 1.f / fmaxf(sqrtf(red[0]), 1e-12f);
  out[(size_t)n * MM + t]       = a * scale;
  out[(size_t)n * MM + t + 256] = b * scale;
}

// x0[n] = [emb[seq[n]], ctx[n]]; also fills lin_in[n][392:904] = ctx[n]
__global__ void build_x0(const float* __restrict__ emb,
                         const int* __restrict__ seq,    // seq + t*BN
                         const float* __restrict__ ctx,
                         float* __restrict__ x0,
                         float* __restrict__ lin_in)
{
  int idx = blockIdx.x * blockDim.x + threadIdx.x;
  if (idx >= BN * KX0) return;
  const int n = idx / KX0, j = idx % KX0;
  if (j < EE) {
    const int tok = seq[n];
    x0[(size_t)n * KX0 + j] = emb[(size_t)tok * EE + j];
  } else {
    const float cv = ctx[(size_t)n * MM + (j - EE)];
    x0[(size_t)n * KX0 + j] = cv;
    lin_in[(size_t)n * KLIN + KX1 + (j - EE)] = cv;
  }
}

// in-place log_softmax over VV per row; one block (256) per row
__global__ __launch_bounds__(256) void log_softmax_row(float* __restrict__ logits)
{
  __shared__ float red[8];
  const int n = blockIdx.x, t = threadIdx.x;
  float* row = logits + (size_t)n * VV;

  float mx = -INFINITY;
  for (int j = t; j < VV; j += 256) mx = fmaxf(mx, row[j]);
#pragma unroll
  for (int off = 16; off; off >>= 1) mx = fmaxf(mx, __shfl_xor(mx, off, 32));
  if ((t & 31) == 0) red[t >> 5] = mx;
  __syncthreads();
  if (t < 8) {
    float x = red[t];
#pragma unroll
    for (int off = 4; off; off >>= 1) x = fmaxf(x, __shfl_xor(x, off, 8));
    if (t == 0) red[0] = x;
  }
  __syncthreads();
  mx = red[0];
  __syncthreads();

  float s = 0.f;
  for (int j = t; j < VV; j += 256) s += __expf(row[j] - mx);
#pragma unroll
  for (int off = 16; off; off >>= 1) s += __shfl_xor(s, off, 32);
  if ((t & 31) == 0) red[t >> 5] = s;
  __syncthreads();
  if (t < 8) {
    float x = red[t];
#pragma unroll
    for (int off = 4; off; off >>= 1) x += __shfl_xor(x, off, 8);
    if (t == 0) red[0] = x;
  }
  __syncthreads();
  const float lse = mx + __logf(red[0]);
  for (int j = t; j < VV; j += 256) row[j] = row[j] - lse;
}

// ==================================================================
extern "C" void kernel_launch(void* const* d_in, const int* in_sizes, int n_in,
                              void* d_out, int out_size, void* d_ws, size_t ws_size,
                              hipStream_t stream)
{
  (void)in_sizes; (void)n_in; (void)out_size; (void)ws_size;

  const float* img    = (const float*)d_in[0];
  const int*   seqs   = (const int*)  d_in[1];
  // d_in[2] sequences_lens: all == T, unused
  const float* conv_w = (const float*)d_in[3];
  const float* conv_b = (const float*)d_in[4];
  const float* fcg_w  = (const float*)d_in[5];
  const float* fcg_b  = (const float*)d_in[6];
  const float* emb    = (const float*)d_in[7];
  const float* w_ih0  = (const float*)d_in[8];
  const float* w_hh0  = (const float*)d_in[9];
  const float* b0     = (const float*)d_in[10];
  const float* w_ih1  = (const float*)d_in[11];
  const float* w_hh1  = (const float*)d_in[12];
  const float* b1     = (const float*)d_in[13];
  const float* lin_w  = (const float*)d_in[14];
  const float* lin_b  = (const float*)d_in[15];
  const float* wp_w   = (const float*)d_in[16];
  const float* wp_b   = (const float*)d_in[17];
  float* out = (float*)d_out;
  float* ws  = (float*)d_ws;

  // ---- workspace layout (floats) ----
  size_t off = 0;
  float* mapped = ws + off; off += (size_t)BN * MM * SS;
  float* pooled = ws + off; off += (size_t)BN * CH;
  float* gvec   = ws + off; off += (size_t)BN * EE;
  float* ctx    = ws + off; off += (size_t)BN * MM;
  float* tdot   = ws + off; off += (size_t)BN * MM;
  float* x0     = ws + off; off += (size_t)BN * KX0;
  float* x1     = ws + off; off += (size_t)BN * KX1;
  float* lin_in = ws + off; off += (size_t)BN * KLIN;
  float* wbuf   = ws + off; off += (size_t)BN * MM;
  float* gates  = ws + off; off += (size_t)2 * BN * G4;
  float* hc     = ws + off; off += (size_t)4 * 2 * BN * EE; // h0,c0,h1,c1
  float* h0 = hc;
  float* c0 = hc + 1 * (size_t)2 * BN * EE;
  float* h1 = hc + 2 * (size_t)2 * BN * EE;
  float* c1 = hc + 3 * (size_t)2 * BN * EE;

  // zero LSTM state (graph-capturable memset node)
  hipMemsetAsync(hc, 0, (size_t)4 * 2 * BN * EE * sizeof(float), stream);

  // ---- prologue ----
  pool_mean<<<(BN * CH + 255) / 256, 256, 0, stream>>>(img, pooled);
  gemm_xwT<<<13, 32, 0, stream>>>(pooled, CH, fcg_w, CH, fcg_b,
                                  gvec, EE, CH, EE);
  conv1x1_gemm<<<dim3(13, 8, BN), 32, 0, stream>>>(img, conv_w, conv_b, mapped);
  attn_dot<<<(BN * MM) / 8, 256, 0, stream>>>(mapped, gvec, tdot);
  l2norm_row<<<BN, 256, 0, stream>>>(tdot, ctx);

  // ---- 24 sequential decoder steps ----
  for (int t = 0; t < TT; ++t) {
    build_x0<<<(BN * KX0 + 255) / 256, 256, 0, stream>>>(emb, seqs + t * BN,
                                                         ctx, x0, lin_in);
    // layer 0 (bidirectional)
    lstm_gates<<<dim3(G4 / 16, 2), 32, 0, stream>>>(
        x0, KX0, h0, w_ih0, w_hh0, b0, gates);
    lstm_cell<<<(2 * BN * EE + 255) / 256, 256, 0, stream>>>(gates, h0, c0,
                                                             x1, KX1);
    // layer 1 (bidirectional); h1 concat goes to lin_in[:, 0:392]
    lstm_gates<<<dim3(G4 / 16, 2), 32, 0, stream>>>(
        x1, KX1, h1, w_ih1, w_hh1, b1, gates);
    lstm_cell<<<(2 * BN * EE + 255) / 256, 256, 0, stream>>>(gates, h1, c1,
                                                             lin_in, KLIN);
    // w = lin_in @ lin_w.T + lin_b
    gemm_xwT<<<MM / 16, 32, 0, stream>>>(
        lin_in, KLIN, lin_w, KLIN, lin_b, wbuf, MM, KLIN, MM);
    // logits = leaky_relu(w) @ wp_w.T + wp_b, then in-place log_softmax
    float* outT = out + (size_t)t * BN * VV;
    wp_gemm<<<VV / 96, 192, 0, stream>>>(wbuf, wp_w, wp_b, outT);
    log_softmax_row<<<BN, 256, 0, stream>>>(outT);
    // next ctx from backward hidden of layer 1
    attn_dot<<<(BN * MM) / 8, 256, 0, stream>>>(mapped, h1 + (size_t)BN * EE,
                                                tdot);
    l2norm_row<<<BN, 256, 0, stream>>>(tdot, ctx);
  }
}